// SpatialGAT_22780506538677
// MI455X (gfx1250) — compile-verified
//
#include <hip/hip_runtime.h>

// ---------------- problem constants ----------------
#define N_  64
#define C_  64
#define T_  64
#define V_  25
#define S_  8
#define W_  3
#define IC_ 8
#define U_  75          // W*V
#define EPS_ 1e-5f

typedef float v2f __attribute__((ext_vector_type(2)));
typedef float v8f __attribute__((ext_vector_type(8)));

// FP32 WMMA: D(16x16) = A(16x4) * B(4x16) + C   -- v_wmma_f32_16x16x4_f32
__device__ __forceinline__ v8f wmma4(v2f a, v2f b, v8f c) {
    return __builtin_amdgcn_wmma_f32_16x16x4_f32(
        /*neg_a=*/false, a, /*neg_b=*/false, b,
        /*c_mod=*/(short)0, c, /*reuse_a=*/false, /*reuse_b=*/false);
}

// ---------------------------------------------------------------------------
// K1: temporal unfold.  x:(N,C,V,T)  ->  upf:(N,C,T,75)  (xp == upf[...,25:50])
// ---------------------------------------------------------------------------
__global__ void __launch_bounds__(256) k_unfold(const float* __restrict__ x,
                                                float* __restrict__ upf) {
    int idx = blockIdx.x * blockDim.x + threadIdx.x;
    if (idx >= N_ * C_ * T_ * U_) return;
    int u = idx % U_; int r = idx / U_;
    int t = r % T_;   r /= T_;
    int c = r % C_;   int n = r / C_;
    int w = u / V_, v = u % V_;
    int tt = t + w - 1;
    float val = 0.f;
    if (tt >= 0 && tt < T_) val = x[((n * C_ + c) * V_ + v) * T_ + tt];
    upf[idx] = val;
}

// ---------------------------------------------------------------------------
// K2: conv1x1 (Cin = O = 64) via WMMA, weights staged in LDS.
// One wave computes the full 64(o) x 16(m) column block: 4 accumulators
// share each B (activation) fragment.
// X element (n,c,m): m -> (t = m/P, v = m%P), addr = (n*64+c)*T*rowLen
//                                                    + t*rowLen + rowOff + v
// ---------------------------------------------------------------------------
__global__ void __launch_bounds__(256) k_conv1x1(
        const float* __restrict__ Wt, const float* __restrict__ bias,
        const float* __restrict__ X, float* __restrict__ Y,
        int M, int P, int rowLen, int rowOff) {
    __shared__ float Wl[64 * 64];
    __shared__ float bl[64];
    for (int i = threadIdx.x; i < 64 * 64; i += 256) Wl[i] = Wt[i];
    if (threadIdx.x < 64) bl[threadIdx.x] = bias[threadIdx.x];
    __syncthreads();

    int wv = (blockIdx.x * blockDim.x + threadIdx.x) >> 5;
    int MT = M / 16;
    if (wv >= N_ * MT) return;
    int n = wv / MT, mt = wv % MT;
    int lane = threadIdx.x & 31, half = lane >> 4, l16 = lane & 15;

    int m = mt * 16 + l16;
    long xoff    = (long)(m / P) * rowLen + rowOff + (m % P);
    long xstride = (long)T_ * rowLen;                 // per-channel plane
    const float* Xn = X + (long)n * 64 * xstride;

    v8f acc[4];
    #pragma unroll
    for (int ot = 0; ot < 4; ++ot) acc[ot] = (v8f){0.f,0.f,0.f,0.f,0.f,0.f,0.f,0.f};

    for (int k0 = 0; k0 < 64; k0 += 4) {
        int kk = k0 + 2 * half;
        v2f b;
        b[0] = Xn[(long)(kk + 0) * xstride + xoff];
        b[1] = Xn[(long)(kk + 1) * xstride + xoff];
        #pragma unroll
        for (int ot = 0; ot < 4; ++ot) {
            v2f a;
            a[0] = Wl[(ot * 16 + l16) * 64 + kk + 0];
            a[1] = Wl[(ot * 16 + l16) * 64 + kk + 1];
            acc[ot] = wmma4(a, b, acc[ot]);
        }
    }
    float* Yn = Y + (long)n * 64 * M;
    #pragma unroll
    for (int ot = 0; ot < 4; ++ot)
        for (int rr = 0; rr < 8; ++rr) {
            int o = ot * 16 + 8 * half + rr;
            Yn[(long)o * M + m] = acc[ot][rr] + bl[o];
        }
}

// ---------------------------------------------------------------------------
// K3: attention scores.  att[n,s,u,v] = (1/512) * sum_{c,t} q[..u]*k[..v]
// One wave handles a u-tile with BOTH v-tiles (shared A fragment).
// q:(N,64,T,75)  k:(N,64,T,25)  att:(N,S,75,25)
// ---------------------------------------------------------------------------
__global__ void __launch_bounds__(256) k_attscore(
        const float* __restrict__ q, const float* __restrict__ kk,
        float* __restrict__ att) {
    const int UT = 5;                         // 75 -> 5x16 (padded)
    int wv = (blockIdx.x * blockDim.x + threadIdx.x) >> 5;
    int total = N_ * S_ * UT;
    if (wv >= total) return;
    int ut = wv % UT; int r = wv / UT;
    int s  = r % S_;  int n = r / S_;
    int lane = threadIdx.x & 31, half = lane >> 4, l16 = lane & 15;
    int u = ut * 16 + l16;
    bool uok = u < U_;
    int va = l16;                 // v-tile 0: always < 25
    int vb = 16 + l16;            // v-tile 1
    bool vbok = vb < V_;

    v8f accA = {0.f,0.f,0.f,0.f,0.f,0.f,0.f,0.f};
    v8f accB = {0.f,0.f,0.f,0.f,0.f,0.f,0.f,0.f};
    for (int c = 0; c < IC_; ++c) {
        int ch = s * IC_ + c;
        const float* qb = q  + (long)(n * 64 + ch) * T_ * U_;
        const float* kb = kk + (long)(n * 64 + ch) * T_ * V_;
        for (int t0 = 0; t0 < T_; t0 += 4) {
            int t = t0 + 2 * half;
            v2f a, b0, b1;
            a[0]  = uok ? qb[(t + 0) * U_ + u] : 0.f;
            a[1]  = uok ? qb[(t + 1) * U_ + u] : 0.f;
            b0[0] = kb[(t + 0) * V_ + va];
            b0[1] = kb[(t + 1) * V_ + va];
            b1[0] = vbok ? kb[(t + 0) * V_ + vb] : 0.f;
            b1[1] = vbok ? kb[(t + 1) * V_ + vb] : 0.f;
            accA = wmma4(a, b0, accA);
            accB = wmma4(a, b1, accB);
        }
    }
    const float invKT = 1.0f / 512.0f;       // 1/(IC*T)
    for (int rr = 0; rr < 8; ++rr) {
        int uu = ut * 16 + 8 * half + rr;
        if (uu < U_) {
            long base = ((long)(n * S_ + s) * U_ + uu) * V_;
            att[base + va] = accA[rr] * invKT;
            if (vbok) att[base + vb] = accB[rr] * invKT;
        }
    }
}

// ---------------------------------------------------------------------------
// K4: d[n,c,t,v] = sum_u xp[n,c,t,u] * graph_a[u,v]   (25-wide, VALU)
// ---------------------------------------------------------------------------
__global__ void __launch_bounds__(256) k_graphmul(
        const float* __restrict__ upf, const float* __restrict__ ga,
        float* __restrict__ d) {
    __shared__ float gs[V_ * V_];
    int nc = blockIdx.x;                                     // n*64 + c
    for (int i = threadIdx.x; i < V_ * V_; i += blockDim.x) gs[i] = ga[i];
    __syncthreads();
    const float* xpb = upf + (long)nc * T_ * U_ + V_;        // xp slice
    float* db = d + (long)nc * T_ * V_;
    for (int i = threadIdx.x; i < T_ * V_; i += blockDim.x) {
        int t = i / V_, v = i % V_;
        const float* xr = xpb + t * U_;
        float sum = 0.f;
        for (int u = 0; u < V_; ++u) sum += xr[u] * gs[u * V_ + v];
        db[i] = sum;
    }
}

// ---------------------------------------------------------------------------
// K5: gate[n,s,w] = sigmoid(mean_{c',t,v} dd[n,s*8+c',t,v] - upf[...,w*25+v])
// ---------------------------------------------------------------------------
__global__ void __launch_bounds__(256) k_gate(
        const float* __restrict__ dd, const float* __restrict__ upf,
        float* __restrict__ gate) {
    __shared__ float sd[256];
    int b = blockIdx.x;                                      // (n*S+s)*W + w
    int w = b % W_; int r = b / W_;
    int s = r % S_; int n = r / S_;
    float sum = 0.f;
    for (int i = threadIdx.x; i < IC_ * T_ * V_; i += blockDim.x) {
        int v = i % V_; int r2 = i / V_;
        int t = r2 % T_; int cp = r2 / T_;
        long ch = (long)(n * C_ + s * IC_ + cp);
        float a  = dd [(ch * T_ + t) * V_ + v];
        float bb = upf[(ch * T_ + t) * U_ + w * V_ + v];
        sum += a - bb;
    }
    sd[threadIdx.x] = sum; __syncthreads();
    for (int st = 128; st > 0; st >>= 1) {
        if (threadIdx.x < st) sd[threadIdx.x] += sd[threadIdx.x + st];
        __syncthreads();
    }
    if (threadIdx.x == 0) {
        float mval = sd[0] / (float)(IC_ * T_ * V_);
        gate[b] = 1.0f / (1.0f + __expf(-mval));
    }
}

// ---------------------------------------------------------------------------
// K6: masked softmax over v, * gate[n,s,u%3], + att0.  One wave32 per row.
// ---------------------------------------------------------------------------
__global__ void __launch_bounds__(256) k_softmax(
        float* __restrict__ att, const float* __restrict__ graph,
        const float* __restrict__ gate, const float* __restrict__ att0) {
    int row = (blockIdx.x * blockDim.x + threadIdx.x) >> 5;
    if (row >= N_ * S_ * U_) return;
    int u = row % U_; int r = row / U_;
    int s = r % S_;   int n = r / S_;
    int lane = threadIdx.x & 31;
    bool ok = lane < V_;

    float val = -3.0e38f;
    if (ok) {
        float raw = att[(long)row * V_ + lane];
        bool m = graph[(s * V_ + (u % V_)) * V_ + lane] > 0.f;
        val = m ? raw : -9.0e15f;
    }
    float mx = val;
    for (int o = 16; o; o >>= 1) mx = fmaxf(mx, __shfl_xor(mx, o, 32));
    float e = ok ? __expf(val - mx) : 0.f;
    float sm = e;
    for (int o = 16; o; o >>= 1) sm += __shfl_xor(sm, o, 32);
    if (ok) {
        float g  = gate[(n * S_ + s) * W_ + (u % W_)];
        float a0 = att0[(s * U_ + u) * V_ + lane];
        att[(long)row * V_ + lane] = (e / sm) * g + a0;
    }
}

// ---------------------------------------------------------------------------
// K7: fused aggregate + out-conv + BN + residual + leaky, per (n,t).
// stage1: M[s*64+c, v] = sum_u upfT[c,u]*att[n,s,u,v]   (WMMA, half of s at a
//         time; each job computes both v-tiles sharing the A fragment)
// stage2: z[o,v] = sum_k out_w[o,k]*M[k,v]              (WMMA, acc over halves)
// y1 = leaky(xp + BN(z + out_b))
// ---------------------------------------------------------------------------
__global__ void __launch_bounds__(256) k_aggregate(
        const float* __restrict__ upf, const float* __restrict__ att,
        const float* __restrict__ out_w, const float* __restrict__ out_b,
        const float* __restrict__ og, const float* __restrict__ ob,
        const float* __restrict__ om, const float* __restrict__ ovv,
        float* __restrict__ y1) {
    __shared__ float upfT[64 * 80];          // 20.5 KB (u padded 75->80)
    __shared__ float Ml[256 * 26];           // 26.6 KB (256 k-rows, stride 26)
    int nt = blockIdx.x;
    int t = nt % T_, n = nt / T_;
    int wave = threadIdx.x >> 5;
    int lane = threadIdx.x & 31, half = lane >> 4, l16 = lane & 15;

    for (int i = threadIdx.x; i < 64 * 80; i += 256) {
        int c = i / 80, u = i % 80;
        upfT[i] = (u < U_) ? upf[((long)(n * C_ + c) * T_ + t) * U_ + u] : 0.f;
    }
    int ot = wave >> 1, vt = wave & 1;       // stage-2 job for this wave
    int vcol = vt * 16 + l16;
    v8f acc2 = {0.f,0.f,0.f,0.f,0.f,0.f,0.f,0.f};
    __syncthreads();

    int va = l16;                 // v-tile 0 col (<25 always)
    int vb = 16 + l16;            // v-tile 1 col
    bool vbok = vb < V_;

    for (int h = 0; h < 2; ++h) {            // two halves of s (4 each)
        // ---- stage 1: fill Ml rows 0..255 for s in [4h, 4h+4) ----
        for (int j = wave; j < 16; j += 8) {
            int ct = j & 3, sL = j >> 2;
            int s = h * 4 + sL;
            const float* ab = att + (long)(n * S_ + s) * U_ * V_;
            v8f accA = {0.f,0.f,0.f,0.f,0.f,0.f,0.f,0.f};
            v8f accB = {0.f,0.f,0.f,0.f,0.f,0.f,0.f,0.f};
            for (int k0 = 0; k0 < 80; k0 += 4) {
                int u0 = k0 + 2 * half;
                bool u0ok = (u0 + 0) < U_, u1ok = (u0 + 1) < U_;
                v2f a, b0, b1;
                a[0]  = upfT[(ct * 16 + l16) * 80 + u0 + 0];
                a[1]  = upfT[(ct * 16 + l16) * 80 + u0 + 1];
                b0[0] = u0ok ? ab[(u0 + 0) * V_ + va] : 0.f;
                b0[1] = u1ok ? ab[(u0 + 1) * V_ + va] : 0.f;
                b1[0] = (vbok && u0ok) ? ab[(u0 + 0) * V_ + vb] : 0.f;
                b1[1] = (vbok && u1ok) ? ab[(u0 + 1) * V_ + vb] : 0.f;
                accA = wmma4(a, b0, accA);
                accB = wmma4(a, b1, accB);
            }
            for (int rr = 0; rr < 8; ++rr) {
                int rrow = sL * 64 + ct * 16 + 8 * half + rr;
                Ml[rrow * 26 + va] = accA[rr];
                if (vbok) Ml[rrow * 26 + vb] = accB[rr];
            }
        }
        __syncthreads();
        // ---- stage 2: partial GEMM over this half's 256 k-rows ----
        {
            bool vok = vcol < V_;
            const float* Wr = out_w + (long)(ot * 16 + l16) * 512 + 256 * h;
            for (int k0 = 0; k0 < 256; k0 += 4) {
                int kk2 = k0 + 2 * half;
                v2f a, b;
                a[0] = Wr[kk2 + 0];
                a[1] = Wr[kk2 + 1];
                b[0] = vok ? Ml[(kk2 + 0) * 26 + vcol] : 0.f;
                b[1] = vok ? Ml[(kk2 + 1) * 26 + vcol] : 0.f;
                acc2 = wmma4(a, b, acc2);
            }
        }
        __syncthreads();
    }
    // ---- epilogue: bias + BN + residual + leaky ----
    if (vcol < V_) {
        for (int rr = 0; rr < 8; ++rr) {
            int o = ot * 16 + 8 * half + rr;
            float val = acc2[rr] + out_b[o];
            float sc = og[o] * rsqrtf(ovv[o] + EPS_);
            val = val * sc + (ob[o] - om[o] * sc);
            float xpv = upfT[o * 80 + V_ + vcol];        // xp slice in LDS
            val = xpv + val;
            val = val > 0.f ? val : 0.1f * val;
            y1[((long)(n * C_ + o) * T_ + t) * V_ + vcol] = val;
        }
    }
}

// ---------------------------------------------------------------------------
// K8: ff-conv WMMA (weights in LDS, 4 o-tiles per wave) + BN + residual
//     + leaky + final ReLU + transpose store.
// out[n,c,v,t] = relu( leaky(xp + BN(ff(y1)+ff_b)) + xp )
// ---------------------------------------------------------------------------
__global__ void __launch_bounds__(256) k_final(
        const float* __restrict__ y1, const float* __restrict__ upf,
        const float* __restrict__ ffw, const float* __restrict__ ffb,
        const float* __restrict__ fg, const float* __restrict__ fb2,
        const float* __restrict__ fm, const float* __restrict__ fv,
        float* __restrict__ out) {
    __shared__ float Wl[64 * 64];
    for (int i = threadIdx.x; i < 64 * 64; i += 256) Wl[i] = ffw[i];
    __syncthreads();

    const int MT = (T_ * V_) / 16;           // 100
    int wv = (blockIdx.x * blockDim.x + threadIdx.x) >> 5;
    if (wv >= N_ * MT) return;
    int n = wv / MT, mt = wv % MT;
    int lane = threadIdx.x & 31, half = lane >> 4, l16 = lane & 15;
    int m = mt * 16 + l16;
    const float* Yn = y1 + (long)n * C_ * T_ * V_;

    v8f acc[4];
    #pragma unroll
    for (int ot = 0; ot < 4; ++ot) acc[ot] = (v8f){0.f,0.f,0.f,0.f,0.f,0.f,0.f,0.f};

    for (int k0 = 0; k0 < C_; k0 += 4) {
        int kk2 = k0 + 2 * half;
        v2f b;
        b[0] = Yn[(long)(kk2 + 0) * T_ * V_ + m];
        b[1] = Yn[(long)(kk2 + 1) * T_ * V_ + m];
        #pragma unroll
        for (int ot = 0; ot < 4; ++ot) {
            v2f a;
            a[0] = Wl[(ot * 16 + l16) * 64 + kk2 + 0];
            a[1] = Wl[(ot * 16 + l16) * 64 + kk2 + 1];
            acc[ot] = wmma4(a, b, acc[ot]);
        }
    }
    int t = m / V_, v = m % V_;
    #pragma unroll
    for (int ot = 0; ot < 4; ++ot)
        for (int rr = 0; rr < 8; ++rr) {
            int o = ot * 16 + 8 * half + rr;
            float val = acc[ot][rr] + ffb[o];
            float sc = fg[o] * rsqrtf(fv[o] + EPS_);
            val = val * sc + (fb2[o] - fm[o] * sc);
            float xpv = upf[((long)(n * C_ + o) * T_ + t) * U_ + V_ + v];
            float yv = xpv + val;
            yv = yv > 0.f ? yv : 0.1f * yv;
            out[((long)(n * C_ + o) * V_ + v) * T_ + t] = fmaxf(yv + xpv, 0.f);
        }
}

// ---------------------------------------------------------------------------
extern "C" void kernel_launch(void* const* d_in, const int* in_sizes, int n_in,
                              void* d_out, int out_size, void* d_ws, size_t ws_size,
                              hipStream_t stream) {
    const float* x      = (const float*)d_in[0];
    const float* graph  = (const float*)d_in[1];
    const float* ga     = (const float*)d_in[2];
    const float* in_w   = (const float*)d_in[3];
    const float* in_b   = (const float*)d_in[4];
    const float* inup_w = (const float*)d_in[5];
    const float* inup_b = (const float*)d_in[6];
    const float* diff_w = (const float*)d_in[7];
    const float* diff_b = (const float*)d_in[8];
    const float* att0   = (const float*)d_in[9];
    const float* out_w  = (const float*)d_in[10];
    const float* out_b  = (const float*)d_in[11];
    const float* og     = (const float*)d_in[12];
    const float* ob     = (const float*)d_in[13];
    const float* om     = (const float*)d_in[14];
    const float* ov     = (const float*)d_in[15];
    const float* ffw    = (const float*)d_in[16];
    const float* ffb    = (const float*)d_in[17];
    const float* fg     = (const float*)d_in[18];
    const float* fb     = (const float*)d_in[19];
    const float* fm     = (const float*)d_in[20];
    const float* fv     = (const float*)d_in[21];

    float* ws   = (float*)d_ws;
    float* upf  = ws;                     // 19,660,800
    float* kb   = ws + 19660800;          //  6,553,600
    float* qb   = ws + 26214400;          // 19,660,800
    float* attb = ws + 45875200;          //    960,000
    float* gateb= ws + 46835200;          //      1,536
    float* db   = qb;                     // reuse (q consumed by attscore)
    float* ddb  = kb;                     // reuse (k consumed by attscore)
    float* y1   = qb + 6553600;           // reuse (second part of q buffer)

    // 1. unfold
    {
        int tot = N_ * C_ * T_ * U_;
        k_unfold<<<(tot + 255) / 256, 256, 0, stream>>>(x, upf);
    }
    // 2. k = in_w * xp + in_b            (X = xp slice of upf)
    {
        int waves = N_ * (1600 / 16);
        k_conv1x1<<<(waves + 7) / 8, 256, 0, stream>>>(
            in_w, in_b, upf, kb, 1600, 25, 75, 25);
    }
    // 3. q = inup_w * upf + inup_b
    {
        int waves = N_ * (4800 / 16);
        k_conv1x1<<<(waves + 7) / 8, 256, 0, stream>>>(
            inup_w, inup_b, upf, qb, 4800, 75, 75, 0);
    }
    // 4. raw attention scores
    {
        int waves = N_ * S_ * 5;
        k_attscore<<<(waves + 7) / 8, 256, 0, stream>>>(qb, kb, attb);
    }
    // 5. d = xp @ graph_a
    k_graphmul<<<N_ * C_, 256, 0, stream>>>(upf, ga, db);
    // 6. dd = diff_w * d + diff_b
    {
        int waves = N_ * (1600 / 16);
        k_conv1x1<<<(waves + 7) / 8, 256, 0, stream>>>(
            diff_w, diff_b, db, ddb, 1600, 25, 25, 0);
    }
    // 7. gate
    k_gate<<<N_ * S_ * W_, 256, 0, stream>>>(ddb, upf, gateb);
    // 8. masked softmax * gate + att0  (in place on attb)
    {
        int rows = N_ * S_ * U_;
        k_softmax<<<(rows + 7) / 8, 256, 0, stream>>>(attb, graph, gateb, att0);
    }
    // 9. aggregate + out conv + BN + residual + leaky  -> y1
    k_aggregate<<<N_ * T_, 256, 0, stream>>>(
        upf, attb, out_w, out_b, og, ob, om, ov, y1);
    // 10. ff conv + BN + residual + leaky + relu + transpose -> out
    {
        int waves = N_ * (1600 / 16);
        k_final<<<(waves + 7) / 8, 256, 0, stream>>>(
            y1, upf, ffw, ffb, fg, fb, fm, fv, (float*)d_out);
    }
    (void)in_sizes; (void)n_in; (void)out_size; (void)ws_size;
}